// ClusteringLoss_6305011990777
// MI455X (gfx1250) — compile-verified
//
#include <hip/hip_runtime.h>

// ---------------------------------------------------------------------------
// ClusteringLoss on MI455X (gfx1250).
// Bandwidth-bound (≈352 MB / 23.3 TB/s ≈ 15 us ideal). Scalar f32 VALU path
// for the pointwise math; V_WMMA_F32_16X16X4_F32 used for exact 32-lane
// reductions (ones-matrix trick, layout-robust); global_prefetch_b8 (WGP
// scope) for the row-ahead stream. All loop-body loads are unconditional
// (boundary handling via clamped addresses) so the hot loop is divergence-
// free straight-line code.
// ---------------------------------------------------------------------------

typedef __attribute__((ext_vector_type(2))) float v2f;
typedef __attribute__((ext_vector_type(8))) float v8f;

static constexpr int Hc  = 1024;
static constexpr int Wc  = 1024;
static constexpr int SEG = 32;           // row segments per (b,k) image

// Exact sum of one f32 value across all 32 lanes of a wave using
// V_WMMA_F32_16X16X4_F32.  Step 1: A = per-lane value (A-layout documented:
// VGPR0 lanes0-15 -> K=0, lanes16-31 -> K=2; VGPR1 = 0), B = ones =>
// D[m][*] = v(m)+v(m+16) = s(m), stored VGPR r: lanes0-15 = s(r),
// lanes16-31 = s(r+8).  Step 2: chain 4 WMMAs with A = ones and B = pairs of
// D-VGPRs; each half-VGPR is value-uniform so every column of B contains the
// 4 distinct s-values under any legal "row within one VGPR" B layout.
// Result: every element of the accumulator = sum over all 32 lanes.
__device__ __forceinline__ float wave_sum32(float v) {
  v2f a;   a[0] = v;    a[1] = 0.0f;
  v2f one; one[0] = 1.0f; one[1] = 1.0f;
  v8f z = {0.f, 0.f, 0.f, 0.f, 0.f, 0.f, 0.f, 0.f};
  v8f d1 = __builtin_amdgcn_wmma_f32_16x16x4_f32(
      false, a, false, one, (short)0, z, false, false);
  v8f acc = {0.f, 0.f, 0.f, 0.f, 0.f, 0.f, 0.f, 0.f};
#pragma unroll
  for (int i = 0; i < 4; ++i) {
    v2f bp; bp[0] = d1[2 * i]; bp[1] = d1[2 * i + 1];
    acc = __builtin_amdgcn_wmma_f32_16x16x4_f32(
        false, one, false, bp, (short)0, acc, false, false);
  }
  return acc[0];
}

// Kernel 1: w_edge(b,h,w) = 1 / (1 + 10 * sqrt(max(gx^2+gy^2, 1e-24)) * min(s))
// gx = forward diff along H (clamped), gy = forward diff along W (clamped).
__global__ __launch_bounds__(256) void edge_w_kernel(
    const float* __restrict__ I, const float* __restrict__ sp,
    float* __restrict__ wout, int B) {
  const int HW4 = Hc * Wc / 4;
  int idx4 = blockIdx.x * blockDim.x + threadIdx.x;
  if (idx4 >= B * HW4) return;
  int b   = idx4 / HW4;
  int rem = idx4 - b * HW4;
  int h   = rem / (Wc / 4);
  int w0  = (rem - h * (Wc / 4)) * 4;

  const float sx = sp[0], sy = sp[1];
  const float mins = fminf(sx, sy);
  const float isx = 1.0f / sx, isy = 1.0f / sy;

  const float* Ib = I + (size_t)b * Hc * Wc;
  const int hd = (h + 1 < Hc) ? h + 1 : Hc - 1;    // clamped down-row
  const int wn = (w0 + 4 < Wc) ? w0 + 4 : Wc - 1;  // clamped right-neighbor
  const float4 cur = *(const float4*)(Ib + (size_t)h  * Wc + w0);
  const float4 dn  = *(const float4*)(Ib + (size_t)hd * Wc + w0);
  const float  nx  = Ib[(size_t)h * Wc + wn];      // == cur.w at w0=Wc-4

  float gx0 = (dn.x - cur.x) * isx, gy0 = (cur.y - cur.x) * isy;
  float gx1 = (dn.y - cur.y) * isx, gy1 = (cur.z - cur.y) * isy;
  float gx2 = (dn.z - cur.z) * isx, gy2 = (cur.w - cur.z) * isy;
  float gx3 = (dn.w - cur.w) * isx, gy3 = (nx    - cur.w) * isy;

  float4 wv;
  wv.x = 1.0f / (1.0f + 10.0f * sqrtf(fmaxf(gx0*gx0 + gy0*gy0, 1e-24f)) * mins);
  wv.y = 1.0f / (1.0f + 10.0f * sqrtf(fmaxf(gx1*gx1 + gy1*gy1, 1e-24f)) * mins);
  wv.z = 1.0f / (1.0f + 10.0f * sqrtf(fmaxf(gx2*gx2 + gy2*gy2, 1e-24f)) * mins);
  wv.w = 1.0f / (1.0f + 10.0f * sqrtf(fmaxf(gx3*gx3 + gy3*gy3, 1e-24f)) * mins);
  *(float4*)(wout + (size_t)b * Hc * Wc + (size_t)h * Wc + w0) = wv;
}

// Kernel 2: per-block row band of one (b,k) image; accumulate nom/denom.
__global__ __launch_bounds__(256) void cut_kernel(
    const float* __restrict__ P, const float* __restrict__ Wm,
    float* __restrict__ acc, int Bn, int Kn) {
  const int rowsPerSeg = Hc / SEG;
  const int seg = blockIdx.x % SEG;
  const int bk  = blockIdx.x / SEG;
  const int b   = bk / Kn;

  const float* pimg = P  + (size_t)bk * Hc * Wc;
  const float* wimg = Wm + (size_t)b  * Hc * Wc;

  const int t  = threadIdx.x;
  const int w0 = t * 4;                             // 256 threads * 4 = Wc
  const int wn = (w0 + 4 < Wc) ? w0 + 4 : Wc - 1;   // clamped, loop-invariant
  const int r0 = seg * rowsPerSeg;

  float4 cur = *(const float4*)(pimg + (size_t)r0 * Wc + w0);
  float nomA = 0.0f, denA = 0.0f;

  for (int h = r0; h < r0 + rowsPerSeg; ++h) {
    const int hd = (h + 1 < Hc) ? h + 1 : Hc - 1;   // scalar min, uniform
    const int hp = (h + 4 < Hc) ? h + 4 : Hc - 1;
    const float4 dn = *(const float4*)(pimg + (size_t)hd * Wc + w0);
    const float  nx = pimg[(size_t)h * Wc + wn];    // == cur.w for last thread
    const float4 wv = *(const float4*)(wimg + (size_t)h * Wc + w0);
    __builtin_prefetch(pimg + (size_t)hp * Wc + w0, 0, 3);  // global_prefetch_b8

    float s0 = (dn.x - cur.x) + (cur.y - cur.x);
    float s1 = (dn.y - cur.y) + (cur.z - cur.y);
    float s2 = (dn.z - cur.z) + (cur.w - cur.z);
    float s3 = (dn.w - cur.w) + (nx    - cur.w);

    nomA = fmaf(cur.x, fmaf(wv.x, s0, cur.x), nomA);
    nomA = fmaf(cur.y, fmaf(wv.y, s1, cur.y), nomA);
    nomA = fmaf(cur.z, fmaf(wv.z, s2, cur.z), nomA);
    nomA = fmaf(cur.w, fmaf(wv.w, s3, cur.w), nomA);

    denA = fmaf(cur.x, fmaf(2.0f, wv.x, 1.0f), denA);
    denA = fmaf(cur.y, fmaf(2.0f, wv.y, 1.0f), denA);
    denA = fmaf(cur.z, fmaf(2.0f, wv.z, 1.0f), denA);
    denA = fmaf(cur.w, fmaf(2.0f, wv.w, 1.0f), denA);

    cur = dn;  // register carry: row h+1 already loaded
  }

  // Intra-wave reduction via WMMA (whole block active -> EXEC all ones).
  float nomW = wave_sum32(nomA);
  float denW = wave_sum32(denA);

  __shared__ float sN[8], sD[8];
  const int lane = t & 31, wid = t >> 5;
  if (lane == 0) { sN[wid] = nomW; sD[wid] = denW; }
  __syncthreads();
  if (t == 0) {
    float n = 0.0f, d = 0.0f;
#pragma unroll
    for (int i = 0; i < 8; ++i) { n += sN[i]; d += sD[i]; }
    unsafeAtomicAdd(&acc[bk], n);                 // global_atomic_add_f32
    unsafeAtomicAdd(&acc[Bn * Kn + bk], d);
  }
}

__global__ void zero_acc(float* acc, int n) {
  int i = blockIdx.x * blockDim.x + threadIdx.x;
  if (i < n) acc[i] = 0.0f;
}

__global__ void finalize_kernel(const float* __restrict__ acc,
                                float* __restrict__ out, int nbk) {
  __shared__ float sc[64];
  int t = threadIdx.x;
  float c = (t < nbk) ? acc[t] / acc[nbk + t] : 0.0f;
  sc[t] = c;
  __syncthreads();
  if (t == 0) {
    float s = 0.0f;
    for (int i = 0; i < 64; ++i) s += sc[i];
    out[0] = (float)nbk - s;
  }
}

extern "C" void kernel_launch(void* const* d_in, const int* in_sizes, int n_in,
                              void* d_out, int out_size, void* d_ws, size_t ws_size,
                              hipStream_t stream) {
  const float* I  = (const float*)d_in[0];   // (B,1,H,W) f32
  const float* sp = (const float*)d_in[1];   // (2,) f32
  const float* P  = (const float*)d_in[2];   // (B,K,H,W) f32
  float* out = (float*)d_out;
  float* ws  = (float*)d_ws;

  const int B = in_sizes[0] / (Hc * Wc);
  const int K = in_sizes[2] / in_sizes[0];
  const int nbk = B * K;

  float* acc  = ws;          // 2*nbk floats
  float* wbuf = ws + 256;    // B*H*W floats, 1 KB offset keeps 16B alignment

  zero_acc<<<1, 256, 0, stream>>>(acc, 2 * nbk);

  const int blocksA = (B * Hc * Wc / 4 + 255) / 256;
  edge_w_kernel<<<blocksA, 256, 0, stream>>>(I, sp, wbuf, B);

  cut_kernel<<<nbk * SEG, 256, 0, stream>>>(P, wbuf, acc, B, K);

  finalize_kernel<<<1, 64, 0, stream>>>(acc, out, nbk);
}